// Prop_29832842838786
// MI455X (gfx1250) — compile-verified
//
#include <hip/hip_runtime.h>
#include <math.h>

typedef __attribute__((ext_vector_type(2))) float v2f;
typedef __attribute__((ext_vector_type(8))) float v8f;

#define NYG 384
#define NXG 384
#define NSH 4
#define NRECV 96
#define NTT 600
#define TILE 16
#define TILES_1D (NYG / TILE)     // 24
#define TILES_TOT (NSH * TILES_1D * TILES_1D)  // 2304
#define WAVES_PER_BLK 4
#define HALO 2
#define LW 20                      // 16 + 2*HALO staged tile width
#define DXC 4.0f
#define DTC 0.0005f
#define PWC 20.0f

// ---------------------------------------------------------------- zero init
__global__ void wave_zero_kernel(float* __restrict__ p, int n) {
    int i = blockIdx.x * blockDim.x + threadIdx.x;
    if (i < n) p[i] = 0.0f;
}

// --------------------------------- per-cell coeffs packed {v2dt2, 1/a, b, 0}
__global__ void wave_aux_kernel(const float* __restrict__ vp,
                                float4* __restrict__ pk) {
    int i = blockIdx.x * blockDim.x + threadIdx.x;
    if (i >= NYG * NXG) return;
    int y = i / NXG, x = i % NXG;
    float d0 = 3.0f * 2500.0f * logf(1000.0f) / (2.0f * PWC * DXC);
    float fy = (float)y, fx = (float)x;
    float myl = fmaxf((PWC - fy) / PWC, 0.0f);
    float myr = fmaxf((fy - (float)(NYG - 1) + PWC) / PWC, 0.0f);
    float mxl = fmaxf((PWC - fx) / PWC, 0.0f);
    float mxr = fmaxf((fx - (float)(NXG - 1) + PWC) / PWC, 0.0f);
    float my = fmaxf(myl, myr), mx = fmaxf(mxl, mxr);
    float sigma = d0 * my * my + d0 * mx * mx;
    float a = 1.0f + 0.5f * DTC * sigma;
    float b = 1.0f - 0.5f * DTC * sigma;
    float v = vp[i];
    float4 c;
    c.x = v * v * (DTC * DTC);   // v^2 dt^2
    c.y = 1.0f / a;              // 1/a
    c.z = b;                     // b
    c.w = 0.0f;
    pk[i] = c;
}

// ---------------------------------------------------------- fused FDTD step
// 4 waves / block, one wave32 per 16x16 tile. Laplacian via 10x
// V_WMMA_F32_16X16X4_F32: ly = A_y(16x20)*Ue(20x16), lx = Ue(16x20)*A_y^T.
// Also gathers receivers for step t-1 from uC (read-only this step).
__global__ __launch_bounds__(32 * WAVES_PER_BLK) void wave_step_kernel(
    float* __restrict__ uP,            // prev in, next out (in-place per cell)
    const float* __restrict__ uC,      // current field (neighborhood reads)
    const float4* __restrict__ pk,     // {v2dt2, 1/a, b, 0}
    const int*   __restrict__ src_loc, // [S,1,2]
    const float* __restrict__ src_amp, // [S,1,NT]
    const int*   __restrict__ rec_loc, // [S,NREC,2]
    float*       __restrict__ out,     // [S,NREC,NT]
    int t) {
    __shared__ float tile[WAVES_PER_BLK][LW * LW];

    // ---- fold in receiver recording for step t-1 (reads uC only)
    int g = blockIdx.x * (32 * WAVES_PER_BLK) + threadIdx.x;
    if (t > 0 && g < NSH * NRECV) {
        int ry = rec_loc[g * 2 + 0];
        int rx = rec_loc[g * 2 + 1];
        int rs = g / NRECV;
        out[(size_t)g * NTT + (t - 1)] =
            uC[(size_t)rs * NYG * NXG + ry * NXG + rx];
    }

    int w    = threadIdx.x >> 5;                  // wave (tile) within block
    int lane = threadIdx.x & 31;
    int tidx = blockIdx.x * WAVES_PER_BLK + w;    // tile index [0, 2304)
    int s    = tidx / (TILES_1D * TILES_1D);
    int rem  = tidx % (TILES_1D * TILES_1D);
    int Y0   = (rem / TILES_1D) * TILE;
    int X0   = (rem % TILES_1D) * TILE;

    const float* us = uC + (size_t)s * NYG * NXG;
    float* lt = tile[w];

    // Stage 20x20 halo tile (zero-fill outside grid = Dirichlet padding).
    for (int i = lane; i < LW * LW; i += 32) {
        int yy = i / LW, xx = i % LW;
        int gy = Y0 - HALO + yy, gx = X0 - HALO + xx;
        float v = 0.0f;
        if (gy >= 0 && gy < NYG && gx >= 0 && gx < NXG) v = us[gy * NXG + gx];
        lt[i] = v;
    }
    __syncthreads();

    int half = lane >> 4;     // K-pair select (A,B); M+8 select (C,D)
    int li   = lane & 15;     // M for A-role, N for B-role

    // Banded 4th-order coefficient fragment: cf[c][v] = w[k - li - 2],
    // k = 4c + 2*half + v. Same values serve as A (ly) and B (lx).
    const float c0 = -1.0f / 12.0f, c1 = 4.0f / 3.0f, c2 = -2.5f;
    float cf[5][2];
#pragma unroll
    for (int c = 0; c < 5; ++c) {
#pragma unroll
        for (int v = 0; v < 2; ++v) {
            int k = 4 * c + 2 * half + v;
            int d = k - li - 2;
            float wv = 0.0f;
            if (d == 0) wv = c2;
            else if (d == 1 || d == -1) wv = c1;
            else if (d == 2 || d == -2) wv = c0;
            cf[c][v] = wv;
        }
    }

    v8f acc = {};
#pragma unroll
    for (int c = 0; c < 5; ++c) {
        int k0 = 4 * c + 2 * half;
        // ly chunk: A = coeffs, B = u rows (Ue[k, n])
        v2f Ac, Bu;
        Ac.x = cf[c][0];
        Ac.y = cf[c][1];
        Bu.x = lt[(k0 + 0) * LW + (li + HALO)];
        Bu.y = lt[(k0 + 1) * LW + (li + HALO)];
        acc = __builtin_amdgcn_wmma_f32_16x16x4_f32(false, Ac, false, Bu,
                                                    (short)0, acc, false, false);
        // lx chunk: A = u columns (Ue[m, k]), B = coeffs
        v2f Au, Bc;
        Au.x = lt[(li + HALO) * LW + (k0 + 0)];
        Au.y = lt[(li + HALO) * LW + (k0 + 1)];
        Bc.x = cf[c][0];
        Bc.y = cf[c][1];
        acc = __builtin_amdgcn_wmma_f32_16x16x4_f32(false, Au, false, Bc,
                                                    (short)0, acc, false, false);
    }

    // Elementwise damped leapfrog + source injection in C/D layout.
    const float inv_dx2 = 1.0f / (DXC * DXC);
    int sy = src_loc[s * 2 + 0];
    int sx = src_loc[s * 2 + 1];
    float amp = src_amp[s * NTT + t];
    float* up = uP + (size_t)s * NYG * NXG;

#pragma unroll
    for (int v = 0; v < 8; ++v) {
        int m = v + 8 * half;
        int y = Y0 + m, x = X0 + li;
        int gi = y * NXG + x;
        float lap   = acc[v] * inv_dx2;
        float ucur  = lt[(m + HALO) * LW + (li + HALO)];
        float uprev = up[gi];
        float4 cc   = pk[gi];                         // one b128 load
        float un = (2.0f * ucur - cc.z * uprev + cc.x * lap) * cc.y;
        if (y == sy && x == sx) un += amp * cc.x;     // cc.x == v2dt2 at src
        up[gi] = un;
    }
}

// ----------------------------------------- trailing receiver gather (t=NT-1)
__global__ void wave_record_kernel(const float* __restrict__ u,
                                   const int* __restrict__ rec_loc,
                                   float* __restrict__ out, int t) {
    int i = blockIdx.x * blockDim.x + threadIdx.x;
    if (i >= NSH * NRECV) return;
    int ry = rec_loc[i * 2 + 0];
    int rx = rec_loc[i * 2 + 1];
    int s  = i / NRECV;
    out[(size_t)i * NTT + t] = u[(size_t)s * NYG * NXG + ry * NXG + rx];
}

extern "C" void kernel_launch(void* const* d_in, const int* in_sizes, int n_in,
                              void* d_out, int out_size, void* d_ws, size_t ws_size,
                              hipStream_t stream) {
    const float* vp      = (const float*)d_in[0];  // [NY,NX]
    const float* src_amp = (const float*)d_in[1];  // [S,1,NT]
    const int*   src_loc = (const int*)d_in[2];    // [S,1,2]
    const int*   rec_loc = (const int*)d_in[3];    // [S,NREC,2]
    float* out = (float*)d_out;                    // [S,NREC,NT]

    float* ws = (float*)d_ws;
    size_t npg = (size_t)NYG * NXG;
    float*  uA = ws;                               // field buffer A
    float*  uB = uA + NSH * npg;                   // field buffer B
    float4* pk = (float4*)(uB + NSH * npg);        // packed coeffs (16B-aligned:
                                                   // 8*npg floats = 16B multiple)

    int nzero = 2 * NSH * (int)npg;
    wave_zero_kernel<<<(nzero + 255) / 256, 256, 0, stream>>>(uA, nzero);
    wave_aux_kernel<<<((int)npg + 255) / 256, 256, 0, stream>>>(vp, pk);

    int nblocks = TILES_TOT / WAVES_PER_BLK;       // 576 blocks of 4 wave32
    for (int t = 0; t < NTT; ++t) {
        float* uPrevNext = (t & 1) ? uB : uA;      // prev in / next out
        float* uCur      = (t & 1) ? uA : uB;
        wave_step_kernel<<<nblocks, 32 * WAVES_PER_BLK, 0, stream>>>(
            uPrevNext, uCur, pk, src_loc, src_amp, rec_loc, out, t);
    }
    // last step's receivers
    float* uLast = ((NTT - 1) & 1) ? uB : uA;
    wave_record_kernel<<<(NSH * NRECV + 127) / 128, 128, 0, stream>>>(
        uLast, rec_loc, out, NTT - 1);
}